// NonTrasition_60825326846162
// MI455X (gfx1250) — compile-verified
//
#include <hip/hip_runtime.h>
#include <hip/hip_bf16.h>

typedef __attribute__((ext_vector_type(2))) float v2f;
typedef __attribute__((ext_vector_type(8))) float v8f;

#define BB   4
#define CIN  64
#define COUT 64
#define NPTS 8192
#define KNN  16

// ---------------------------------------------------------------------------
// Kernel A: z[b,o,n] = sum_c W[o,c] * x[b,c,n]   via V_WMMA_F32_16X16X4_F32
// One wave -> one 16(M=o) x 16(N=n) tile, looping K(c)=64 in steps of 4.
// Block = 256 threads = 8 waves -> 16 x 128 strip per block.
// ---------------------------------------------------------------------------
__global__ __launch_bounds__(256)
void conv1x1_wmma_kernel(const float* __restrict__ W,
                         const float* __restrict__ x,
                         float* __restrict__ z) {
    const int lane = threadIdx.x & 31;
    const int wave = threadIdx.x >> 5;
    const int half = lane >> 4;      // 0: lanes 0-15, 1: lanes 16-31
    const int l15  = lane & 15;

    const int n0 = blockIdx.x * 128 + wave * 16 + l15; // this lane's N column
    const int o0 = blockIdx.y * 16;                    // M tile base (output ch)
    const int b  = blockIdx.z;

    // A fragment (16x4 f32): lane m=l15 holds A[m][2*half + {0,1}] in v0,v1
    const float* Wrow = W + (size_t)(o0 + l15) * CIN;
    // B fragment (4x16 f32): lane n=l15 holds B[2*half + {0,1}][n] in v0,v1
    const float* Xcol = x + (size_t)b * CIN * NPTS + n0;

    v8f acc = {};
#pragma unroll
    for (int c = 0; c < CIN; c += 4) {
        v2f a, bf;
        a.x  = Wrow[c + 2 * half + 0];
        a.y  = Wrow[c + 2 * half + 1];
        bf.x = Xcol[(size_t)(c + 2 * half + 0) * NPTS];
        bf.y = Xcol[(size_t)(c + 2 * half + 1) * NPTS];
        // (neg_a, A, neg_b, B, c_mod, C, reuse_a, reuse_b)
        acc = __builtin_amdgcn_wmma_f32_16x16x4_f32(
            false, a, false, bf, (short)0, acc, false, false);
    }

    // C/D layout: VGPR v -> row M = v + 8*half, N = l15
    float* Zb = z + (size_t)b * COUT * NPTS + n0;
#pragma unroll
    for (int v = 0; v < 8; ++v) {
        const int o = o0 + v + 8 * half;
        Zb[(size_t)o * NPTS] = acc[v];
    }
}

// ---------------------------------------------------------------------------
// Kernel B: fused KNN (top-16 smallest squared distances) + neighbor-max of z.
// One block = 256 threads = 256 rows of one batch. All 3*8192 coords of the
// batch staged in LDS (96 KB of the 320 KB WGP LDS); candidate scan is an
// LDS broadcast (all lanes read the same j).
// ---------------------------------------------------------------------------
__global__ __launch_bounds__(256)
void knn_gathermax_kernel(const float* __restrict__ coords,
                          const float* __restrict__ z,
                          float* __restrict__ y) {
    extern __shared__ float sc[]; // [0,N): x  [N,2N): y  [2N,3N): z
    const int b = blockIdx.y;
    const float* cb = coords + (size_t)b * 3 * NPTS;

    for (int t = threadIdx.x; t < 3 * NPTS; t += 256) sc[t] = cb[t];
    __syncthreads();

    const int i = blockIdx.x * 256 + threadIdx.x;
    const float xi = sc[i];
    const float yi = sc[NPTS + i];
    const float zi = sc[2 * NPTS + i];

    // register-resident sorted top-16 (ascending distance)
    float bestd[KNN];
    int   besti[KNN];
#pragma unroll
    for (int t = 0; t < KNN; ++t) { bestd[t] = 3.0e38f; besti[t] = 0; }

    for (int j = 0; j < NPTS; ++j) {
        const float dx = xi - sc[j];
        const float dy = yi - sc[NPTS + j];
        const float dz = zi - sc[2 * NPTS + j];
        const float d  = fmaf(dx, dx, fmaf(dy, dy, dz * dz));
        if (d < bestd[KNN - 1]) {
            float cd = d; int ci = j;
#pragma unroll
            for (int t = 0; t < KNN; ++t) {
                if (cd < bestd[t]) {
                    float td = bestd[t]; bestd[t] = cd; cd = td;
                    int   ti = besti[t]; besti[t] = ci; ci = ti;
                }
            }
        }
    }

    // y[b,o,i] = max_k z[b,o,idx[k]]  (z tile lives in L2: 8 MB << 192 MB)
    const float* zb = z + (size_t)b * COUT * NPTS;
    float* yb = y + (size_t)b * COUT * NPTS;
    for (int o = 0; o < COUT; ++o) {
        const float* zo = zb + (size_t)o * NPTS;
        float m = -3.0e38f;
#pragma unroll
        for (int t = 0; t < KNN; ++t) m = fmaxf(m, zo[besti[t]]);
        yb[(size_t)o * NPTS + i] = m;
    }
}

// ---------------------------------------------------------------------------
extern "C" void kernel_launch(void* const* d_in, const int* in_sizes, int n_in,
                              void* d_out, int out_size, void* d_ws, size_t ws_size,
                              hipStream_t stream) {
    const float* x      = (const float*)d_in[0]; // [B, CIN, N]
    const float* coords = (const float*)d_in[1]; // [B, 3, N]
    const float* W      = (const float*)d_in[2]; // [COUT, CIN]
    float* y = (float*)d_out;                    // [B, COUT, N] then coords
    float* z = (float*)d_ws;                     // [B, COUT, N] scratch (8 MB)

    // Kernel A: z = W @ x
    dim3 gA(NPTS / 128, COUT / 16, BB);
    conv1x1_wmma_kernel<<<gA, 256, 0, stream>>>(W, x, z);

    // Kernel B: fused KNN + gather-max
    dim3 gB(NPTS / 256, BB);
    knn_gathermax_kernel<<<gB, 256, 3 * NPTS * sizeof(float), stream>>>(coords, z, y);

    // Second tuple output: coords passthrough
    hipMemcpyAsync(y + (size_t)BB * COUT * NPTS, coords,
                   (size_t)BB * 3 * NPTS * sizeof(float),
                   hipMemcpyDeviceToDevice, stream);
}